// BiSTGNNv2_54829552501060
// MI455X (gfx1250) — compile-verified
//
#include <hip/hip_runtime.h>
#include <math.h>

// ---------------------------------------------------------------------------
// BiSTGNN forward for MI455X (gfx1250). All heavy GEMMs use
// V_WMMA_F32_16X16X4_F32 (full fp32 precision, wave32, 16x16 C tile / wave).
// ---------------------------------------------------------------------------

typedef __attribute__((ext_vector_type(2))) float v2f;
typedef __attribute__((ext_vector_type(8))) float v8f;

#define Bq 32
#define Nq 716
#define Tq 168
#define Lq 64
#define NTP 896          // N+T=884 padded to 16
#define NT 884
#define GATE3N 2148      // 3*716
#define GATE3L 192       // 3*64
#define DIN 200          // T + SED
#define GIN 752          // N + SED + TED

__device__ __forceinline__ v8f wmma4(v2f a, v2f b, v8f c) {
  // D = A(16x4,f32) x B(4x16,f32) + C(16x16,f32)
  return __builtin_amdgcn_wmma_f32_16x16x4_f32(false, a, false, b, (short)0, c,
                                               false, false);
}

__device__ __forceinline__ float eluf(float x) { return x > 0.f ? x : expm1f(x); }
__device__ __forceinline__ float sigf(float x) { return 1.f / (1.f + expf(-x)); }

// ---------------------------------------------------------------------------
// Generic batched WMMA GEMM: C[b] = act(A[b](MxK) @ Bw[b](KxN) + bias)
// block = 128 threads (4 waves); wave w computes 16x16 tile at (m0, n_blk+16w).
// K must be a multiple of 4 (true for every call: 200,752,64,168,716).
// ---------------------------------------------------------------------------
__global__ void __launch_bounds__(128)
gemm_wmma_f32(const float* __restrict__ A, const float* __restrict__ Bw,
              const float* __restrict__ bias, float* __restrict__ C,
              int M, int N, int K,
              long long sA, long long sB, long long sC, int act)
{
  const int bz   = blockIdx.z;
  const float* Ab = A  + (long long)bz * sA;
  const float* Bb = Bw + (long long)bz * sB;
  float*       Cb = C  + (long long)bz * sC;

  const int lane = threadIdx.x & 31;
  const int wave = threadIdx.x >> 5;
  const int half = lane >> 4;      // 0 | 1
  const int lid  = lane & 15;
  const int m0   = blockIdx.x * 16;
  const int n0   = blockIdx.y * 64 + wave * 16;

  int m_l = m0 + lid; if (m_l > M - 1) m_l = M - 1;   // clamp loads
  int n_l = n0 + lid; if (n_l > N - 1) n_l = N - 1;

  v8f acc = {0.f, 0.f, 0.f, 0.f, 0.f, 0.f, 0.f, 0.f};
  for (int k = 0; k < K; k += 4) {
    const int kb = k + 2 * half;
    v2f a, b;
    a.x = Ab[(long long)m_l * K + kb];
    a.y = Ab[(long long)m_l * K + kb + 1];
    b.x = Bb[(long long)kb * N + n_l];
    b.y = Bb[(long long)(kb + 1) * N + n_l];
    acc = wmma4(a, b, acc);
  }

  const float bn = bias ? bias[n_l] : 0.f;
  const bool ncol_ok = (n0 + lid) < N;
#pragma unroll
  for (int v = 0; v < 8; ++v) {
    const int mg = m0 + v + 8 * half;
    if (mg < M && ncol_ok) {
      float val = acc[v] + bn;
      if (act == 1) val = eluf(val);
      Cb[(long long)mg * N + (n0 + lid)] = val;
    }
  }
}

// ---------------------------------------------------------------------------
// Fused GCN message-passing: Out = adj @ Hin, adj = tanh(relu(X X^T) + I),
// adj recomputed tile-by-tile in registers (never materialized: saves 100 MB).
// grid = (56 m-tiles, 1, B), 4 waves; wave w owns feature cols [16w,16w+16).
// ---------------------------------------------------------------------------
__global__ void __launch_bounds__(128)
gcn_adj_mm(const float* __restrict__ Xadj, const float* __restrict__ Hin,
           float* __restrict__ Out)
{
  __shared__ float sS[4][16][16];
  const int bz = blockIdx.z;
  const float* Xa = Xadj + (long long)bz * (NTP * Lq);
  const float* Hb = Hin  + (long long)bz * (NTP * Lq);
  float*       Ob = Out  + (long long)bz * (NTP * Lq);

  const int lane = threadIdx.x & 31;
  const int wave = threadIdx.x >> 5;
  const int half = lane >> 4;
  const int lid  = lane & 15;
  const int m0   = blockIdx.x * 16;
  const int f0   = wave * 16;

  v8f acc = {0.f, 0.f, 0.f, 0.f, 0.f, 0.f, 0.f, 0.f};

  for (int mmt = 0; mmt < NTP / 16; ++mmt) {
    const int mm0 = mmt * 16;
    // ---- S = X[m-tile] @ X[mm-tile]^T  (K = 64) -------------------------
    v8f s = {0.f, 0.f, 0.f, 0.f, 0.f, 0.f, 0.f, 0.f};
    for (int k = 0; k < Lq; k += 4) {
      const int kb = k + 2 * half;
      v2f a, b;
      a.x = Xa[(long long)(m0  + lid) * Lq + kb];
      a.y = Xa[(long long)(m0  + lid) * Lq + kb + 1];
      b.x = Xa[(long long)(mm0 + lid) * Lq + kb];       // B[k][n]=X[mm0+n][k]
      b.y = Xa[(long long)(mm0 + lid) * Lq + kb + 1];
      s = wmma4(a, b, s);
    }
    __syncthreads();                 // protect LDS reuse across iterations
    const int ng = mm0 + lid;
#pragma unroll
    for (int v = 0; v < 8; ++v) {
      const int mg = m0 + v + 8 * half;
      float val = s[v];
      val = val > 0.f ? val : 0.f;                  // relu
      if (mg == ng && ng < NT) val += 1.0f;         // + I (true diagonal only)
      val = tanhf(val);
      if (ng >= NT) val = 0.f;                      // mask padded columns
      sS[wave][v + 8 * half][lid] = val;            // sS[m][n]
    }
    __syncthreads();
    // ---- acc += S(16x16) @ Hin[mm-tile][f-cols] -------------------------
#pragma unroll
    for (int kk = 0; kk < 4; ++kk) {
      const int kb = kk * 4 + 2 * half;
      v2f a, b;
      a.x = sS[wave][lid][kb];
      a.y = sS[wave][lid][kb + 1];
      b.x = Hb[(long long)(mm0 + kb)     * Lq + f0 + lid];
      b.y = Hb[(long long)(mm0 + kb + 1) * Lq + f0 + lid];
      acc = wmma4(a, b, acc);
    }
  }
#pragma unroll
  for (int v = 0; v < 8; ++v) {
    const int mg = m0 + v + 8 * half;
    Ob[(long long)mg * Lq + f0 + lid] = acc[v];     // pad rows: harmless
  }
}

// ---------------------------------------------------------------------------
// Packing / transpose / fill helpers
// ---------------------------------------------------------------------------
__global__ void pack_se_in(const float* __restrict__ x,
                           const float* __restrict__ se_emb,
                           float* __restrict__ out)
{
  long long i = (long long)blockIdx.x * 256 + threadIdx.x;
  const long long total = (long long)Bq * Nq * DIN;
  if (i >= total) return;
  const int col = (int)(i % DIN);
  const long long row = i / DIN;           // b*716 + n
  const int n = (int)(row % Nq);
  out[i] = (col < Tq) ? x[row * Tq + col] : se_emb[n * 32 + (col - Tq)];
}

__global__ void pack_te_in(const float* __restrict__ x,
                           const float* __restrict__ xmark,
                           const float* __restrict__ te_emb,
                           float* __restrict__ out)
{
  long long i = (long long)blockIdx.x * 256 + threadIdx.x;
  const long long total = (long long)Bq * Tq * GIN;
  if (i >= total) return;
  const int col = (int)(i % GIN);
  const long long row = i / GIN;           // b*168 + t
  const int t = (int)(row % Tq);
  const int b = (int)(row / Tq);
  float v;
  if (col < Nq)            v = x[((long long)b * Nq + col) * Tq + t];
  else if (col < Nq + 4)   v = xmark[((long long)b * Tq + t) * 4 + (col - Nq)];
  else                     v = te_emb[t * 32 + (col - Nq - 4)];
  out[i] = v;
}

__global__ void transpose2d(const float* __restrict__ in, float* __restrict__ out,
                            int rows, int cols)     // out[cols][rows]
{
  long long i = (long long)blockIdx.x * 256 + threadIdx.x;
  if (i >= (long long)rows * cols) return;
  const int c = (int)(i % cols);
  const long long r = i / cols;
  out[(long long)c * rows + r] = in[i];
}

__global__ void fill_zero(float* __restrict__ p, long long n)
{
  long long i = (long long)blockIdx.x * 256 + threadIdx.x;
  if (i < n) p[i] = 0.f;
}

__global__ void zero_xpad_tail(float* __restrict__ Xpad)
{
  int i = blockIdx.x * 256 + threadIdx.x;
  if (i >= Bq * 12 * Lq) return;
  const int b = i / (12 * Lq);
  const int r = i % (12 * Lq);
  Xpad[(long long)b * (NTP * Lq) + NT * Lq + r] = 0.f;
}

// ---------------------------------------------------------------------------
// te-branch GRU (hidden 64): one block per batch, full 168-step scan in LDS.
// xw already includes bih; h_t written straight into X_pad rows 716+t.
// ---------------------------------------------------------------------------
__global__ void __launch_bounds__(192)
te_gru_scan(const float* __restrict__ xw,    // (B*168) x 192
            const float* __restrict__ Whh,   // 192 x 64
            const float* __restrict__ bhh,   // 192
            float* __restrict__ Xpad)
{
  __shared__ float h[Lq];
  __shared__ float gh[GATE3L];
  const int b = blockIdx.x;
  const int g = threadIdx.x;
  if (g < Lq) h[g] = 0.f;
  __syncthreads();
  for (int t = 0; t < Tq; ++t) {
    float acc = bhh[g];
    for (int k = 0; k < Lq; ++k) acc += Whh[g * Lq + k] * h[k];
    gh[g] = acc;
    __syncthreads();
    if (g < Lq) {
      const float* xwt = xw + ((long long)b * Tq + t) * GATE3L;
      const float r  = sigf(xwt[g]           + gh[g]);
      const float z  = sigf(xwt[Lq + g]      + gh[Lq + g]);
      const float nn = tanhf(xwt[2 * Lq + g] + r * gh[2 * Lq + g]);
      const float hnew = (1.f - z) * nn + z * h[g];
      h[g] = hnew;
      Xpad[((long long)b * NTP + Nq + t) * Lq + g] = hnew;
    }
    __syncthreads();
  }
}

// ---------------------------------------------------------------------------
// tr-branch GRU gate combine for one timestep (gh precomputed by WMMA GEMM).
// In-place h update is safe (gh already read old h). Writes h into tout[:, :, t].
// ---------------------------------------------------------------------------
__global__ void tr_gru_combine(const float* __restrict__ xw,   // B x 168 x 2148
                               const float* __restrict__ gh,   // 32 x 2148
                               float* __restrict__ h,          // 32 x 716
                               float* __restrict__ tout,       // B x 716 x 168
                               int t)
{
  const int i = blockIdx.x * 256 + threadIdx.x;
  if (i >= Bq * Nq) return;
  const int b = i / Nq, j = i % Nq;
  const float* xwt = xw + ((long long)b * Tq + t) * GATE3N;
  const float* ghb = gh + (long long)b * GATE3N;
  const float r  = sigf(xwt[j]          + ghb[j]);
  const float z  = sigf(xwt[Nq + j]     + ghb[Nq + j]);
  const float nn = tanhf(xwt[2 * Nq + j] + r * ghb[2 * Nq + j]);
  const float hnew = (1.f - z) * nn + z * h[i];
  h[i] = hnew;
  tout[((long long)b * Nq + j) * Tq + t] = hnew;
}

// ---------------------------------------------------------------------------
// Causal dilated TCN (dil 1,2,4) + 1x1 head, evaluated only at t = T-1.
// Receptive field is the last 8 time columns. One thread per (b, n).
// ---------------------------------------------------------------------------
__global__ void tcn_out_kernel(const float* __restrict__ n_out,
                               const float* __restrict__ tout,
                               const float* __restrict__ x,
                               const float* __restrict__ W0, const float* __restrict__ b0,
                               const float* __restrict__ W1, const float* __restrict__ b1,
                               const float* __restrict__ W2, const float* __restrict__ b2,
                               const float* __restrict__ Wo, const float* __restrict__ bo,
                               float* __restrict__ out)
{
  const int i = blockIdx.x * 256 + threadIdx.x;
  if (i >= Bq * Nq) return;
  const long long base = (long long)i * Tq + (Tq - 8);
  float in[3][8];
#pragma unroll
  for (int t = 0; t < 8; ++t) {
    in[0][t] = n_out[base + t];
    in[1][t] = tout[base + t];
    in[2][t] = x[base + t];
  }
  // h0 at t in {161,163,165,167}  (local 1,3,5,7); taps in(t-1), in(t)
  float h0[16][4];
#pragma unroll
  for (int o = 0; o < 16; ++o)
#pragma unroll
    for (int q = 0; q < 4; ++q) {
      const int tt = 1 + 2 * q;
      float a = b0[o];
#pragma unroll
      for (int c = 0; c < 3; ++c)
        a += W0[o * 6 + c * 2] * in[c][tt - 1] + W0[o * 6 + c * 2 + 1] * in[c][tt];
      h0[o][q] = eluf(a);
    }
  // h1 at t in {163,167}; dilation 2 -> taps h0(t-2), h0(t)
  float h1[16][2];
#pragma unroll
  for (int o = 0; o < 16; ++o)
#pragma unroll
    for (int p = 0; p < 2; ++p) {
      float a = b1[o];
#pragma unroll
      for (int c = 0; c < 16; ++c)
        a += W1[o * 32 + c * 2] * h0[c][2 * p] + W1[o * 32 + c * 2 + 1] * h0[c][2 * p + 1];
      h1[o][p] = eluf(a);
    }
  // h2 at t=167; dilation 4 -> taps h1(163), h1(167)
  float h2[16];
#pragma unroll
  for (int o = 0; o < 16; ++o) {
    float a = b2[o];
#pragma unroll
    for (int c = 0; c < 16; ++c)
      a += W2[o * 32 + c * 2] * h1[c][0] + W2[o * 32 + c * 2 + 1] * h1[c][1];
    h2[o] = eluf(a);
  }
  float o = bo[0];
#pragma unroll
  for (int c = 0; c < 16; ++c) o += Wo[c] * h2[c];
  out[i] = o;
}

// ---------------------------------------------------------------------------
// Host orchestration
// ---------------------------------------------------------------------------
static inline int cdiv(int a, int b) { return (a + b - 1) / b; }

extern "C" void kernel_launch(void* const* d_in, const int* in_sizes, int n_in,
                              void* d_out, int out_size, void* d_ws, size_t ws_size,
                              hipStream_t stream)
{
  const float* x        = (const float*)d_in[0];
  const float* xmark    = (const float*)d_in[1];
  const float* se_emb   = (const float*)d_in[2];
  const float* se_W1    = (const float*)d_in[3];
  const float* se_b1    = (const float*)d_in[4];
  const float* se_W2    = (const float*)d_in[5];
  const float* se_b2    = (const float*)d_in[6];
  const float* te_emb   = (const float*)d_in[7];
  const float* te_Wih   = (const float*)d_in[8];
  const float* te_Whh   = (const float*)d_in[9];
  const float* te_bih   = (const float*)d_in[10];
  const float* te_bhh   = (const float*)d_in[11];
  const float* gcn_W    = (const float*)d_in[12];
  const float* gcn_b    = (const float*)d_in[13];
  const float* fr_W1    = (const float*)d_in[14];
  const float* fr_b1    = (const float*)d_in[15];
  const float* fr_W2    = (const float*)d_in[16];
  const float* fr_W2b   = (const float*)d_in[17];
  const float* tr_Wih   = (const float*)d_in[18];
  const float* tr_Whh   = (const float*)d_in[19];
  const float* tr_bih   = (const float*)d_in[20];
  const float* tr_bhh   = (const float*)d_in[21];
  const float* tcn_W0   = (const float*)d_in[22];
  const float* tcn_b0   = (const float*)d_in[23];
  const float* tcn_W1   = (const float*)d_in[24];
  const float* tcn_b1   = (const float*)d_in[25];
  const float* tcn_W2   = (const float*)d_in[26];
  const float* tcn_b2   = (const float*)d_in[27];
  const float* out_W    = (const float*)d_in[28];
  const float* out_b    = (const float*)d_in[29];
  float* out = (float*)d_out;

  // ---- workspace layout (floats) ----
  float* ws = (float*)d_ws;
  size_t off = 0;
  auto alloc = [&](size_t n) { float* p = ws + off; off += (n + 63) & ~(size_t)63; return p; };
  float* WteT    = alloc((size_t)GIN * GATE3L);        // 752 x 192
  float* WtrIhT  = alloc((size_t)Lq * GATE3N);         // 64 x 2148
  float* WtrHhT  = alloc((size_t)Nq * GATE3N);         // 716 x 2148
  float* se_in   = alloc((size_t)Bq * Nq * DIN);
  float* t1      = alloc((size_t)Bq * Nq * Lq);
  float* Xpad    = alloc((size_t)Bq * NTP * Lq);
  float* Xtmp    = alloc((size_t)Bq * NTP * Lq);
  float* AH      = alloc((size_t)Bq * NTP * Lq);
  float* te_in   = alloc((size_t)Bq * Tq * GIN);
  float* xw_te   = alloc((size_t)Bq * Tq * GATE3L);
  float* xw_tr   = alloc((size_t)Bq * Tq * GATE3N);
  float* gh_tr   = alloc((size_t)Bq * GATE3N);
  float* h_tr    = alloc((size_t)Bq * Nq);
  float* t1f     = alloc((size_t)Bq * Nq * Tq);
  float* n_outb  = alloc((size_t)Bq * Nq * Tq);
  float* toutb   = alloc((size_t)Bq * Nq * Tq);
  (void)ws_size; (void)n_in; (void)in_sizes; (void)out_size;

  // ---- weight transposes (one-off, tiny) ----
  transpose2d<<<cdiv(GATE3L * GIN, 256), 256, 0, stream>>>(te_Wih, WteT, GATE3L, GIN);
  transpose2d<<<cdiv(GATE3N * Lq, 256), 256, 0, stream>>>(tr_Wih, WtrIhT, GATE3N, Lq);
  transpose2d<<<cdiv(GATE3N * Nq, 256), 256, 0, stream>>>(tr_Whh, WtrHhT, GATE3N, Nq);

  // ---- SE branch: Xs = elu(se_in@W1+b1)@W2+b2 -> X_pad rows [0,716) ----
  pack_se_in<<<cdiv(Bq * Nq * DIN, 256), 256, 0, stream>>>(x, se_emb, se_in);
  gemm_wmma_f32<<<dim3(Bq * Nq / 16, 1, 1), 128, 0, stream>>>(
      se_in, se_W1, se_b1, t1, Bq * Nq, Lq, DIN, 0, 0, 0, 1);
  gemm_wmma_f32<<<dim3(cdiv(Nq, 16), 1, Bq), 128, 0, stream>>>(
      t1, se_W2, se_b2, Xpad, Nq, Lq, Lq,
      (long long)Nq * Lq, 0, (long long)NTP * Lq, 0);
  zero_xpad_tail<<<cdiv(Bq * 12 * Lq, 256), 256, 0, stream>>>(Xpad);

  // ---- TE branch GRU -> X_pad rows [716,884) ----
  pack_te_in<<<cdiv(Bq * Tq * GIN, 256), 256, 0, stream>>>(x, xmark, te_emb, te_in);
  gemm_wmma_f32<<<dim3(Bq * Tq / 16, GATE3L / 64, 1), 128, 0, stream>>>(
      te_in, WteT, te_bih, xw_te, Bq * Tq, GATE3L, GIN, 0, 0, 0, 0);
  te_gru_scan<<<Bq, 192, 0, stream>>>(xw_te, te_Whh, te_bhh, Xpad);

  // ---- GCN: 2 outer layers x 2 inner hops, adjacency fused ----
  for (int i = 0; i < 2; ++i) {
    const float* W0g = gcn_W + (size_t)(i * 2 + 0) * Lq * Lq;
    const float* W1g = gcn_W + (size_t)(i * 2 + 1) * Lq * Lq;
    const float* b0g = gcn_b + (size_t)(i * 2 + 0) * Lq;
    const float* b1g = gcn_b + (size_t)(i * 2 + 1) * Lq;
    gcn_adj_mm<<<dim3(NTP / 16, 1, Bq), 128, 0, stream>>>(Xpad, Xpad, AH);
    gemm_wmma_f32<<<dim3(cdiv(NT, 16), 1, Bq), 128, 0, stream>>>(
        AH, W0g, b0g, Xtmp, NT, Lq, Lq,
        (long long)NTP * Lq, 0, (long long)NTP * Lq, 1);
    gcn_adj_mm<<<dim3(NTP / 16, 1, Bq), 128, 0, stream>>>(Xpad, Xtmp, AH);
    gemm_wmma_f32<<<dim3(cdiv(NT, 16), 1, Bq), 128, 0, stream>>>(
        AH, W1g, b1g, Xpad, NT, Lq, Lq,
        (long long)NTP * Lq, 0, (long long)NTP * Lq, 1);
    // X_pad pad rows [884,896) stay zero (masked stores) -> adj masking valid
  }

  // ---- fr MLP on Xs = X_pad[:, :716] ----
  gemm_wmma_f32<<<dim3(cdiv(Nq, 16), cdiv(Tq, 64), Bq), 128, 0, stream>>>(
      Xpad, fr_W1, fr_b1, t1f, Nq, Tq, Lq,
      (long long)NTP * Lq, 0, (long long)Nq * Tq, 1);
  gemm_wmma_f32<<<dim3(cdiv(Nq, 16), cdiv(Tq, 64), Bq), 128, 0, stream>>>(
      t1f, fr_W2, fr_W2b, n_outb, Nq, Tq, Tq,
      (long long)Nq * Tq, 0, (long long)Nq * Tq, 0);

  // ---- tr GRU on Xt = X_pad[:, 716:884], hidden 716 ----
  gemm_wmma_f32<<<dim3(cdiv(Tq, 16), cdiv(GATE3N, 64), Bq), 128, 0, stream>>>(
      Xpad + (size_t)Nq * Lq, WtrIhT, tr_bih, xw_tr, Tq, GATE3N, Lq,
      (long long)NTP * Lq, 0, (long long)Tq * GATE3N, 0);
  fill_zero<<<cdiv(Bq * Nq, 256), 256, 0, stream>>>(h_tr, (long long)Bq * Nq);
  for (int t = 0; t < Tq; ++t) {
    gemm_wmma_f32<<<dim3(Bq / 16, cdiv(GATE3N, 64), 1), 128, 0, stream>>>(
        h_tr, WtrHhT, tr_bhh, gh_tr, Bq, GATE3N, Nq, 0, 0, 0, 0);
    tr_gru_combine<<<cdiv(Bq * Nq, 256), 256, 0, stream>>>(xw_tr, gh_tr, h_tr,
                                                           toutb, t);
  }

  // ---- causal TCN head, last timestep only ----
  tcn_out_kernel<<<cdiv(Bq * Nq, 256), 256, 0, stream>>>(
      n_outb, toutb, x, tcn_W0, tcn_b0, tcn_W1, tcn_b1, tcn_W2, tcn_b2,
      out_W, out_b, out);
}